// SSD_32813550142121
// MI455X (gfx1250) — compile-verified
//
#include <hip/hip_runtime.h>
#include <math.h>

// SSD detection post-processing for MI455X (gfx1250, wave32).
// ws layout: [ topk: 32*1500*5 f32 ][ mask: 32*1500*48 u32 ][ act: 32*48 u32 ] ~10MB (L2-resident).

#define NB 32
#define NP 76800
#define TK 1500
#define MASKW 48          // 47 words of 32 bits cover 1500 entries, +1 pad (48 = 16*3 for WMMA scan)
#define CONF_T 0.01f
#define NMS_T 0.2f

typedef unsigned int u32;
typedef unsigned long long u64;
typedef float v2f __attribute__((ext_vector_type(2)));
typedef float v8f __attribute__((ext_vector_type(8)));

#ifndef __has_builtin
#define __has_builtin(x) 0
#endif

__device__ __forceinline__ u32 ballot32(bool p) {
#if __has_builtin(__builtin_amdgcn_ballot_w32)
  return __builtin_amdgcn_ballot_w32(p);
#else
  return (u32)__ballot(p);
#endif
}

#if __has_builtin(__builtin_amdgcn_global_load_async_to_lds_b32) && __has_builtin(__builtin_amdgcn_s_wait_asynccnt)
#define HAVE_ASYNC_LDS 1
typedef __attribute__((address_space(1))) int gbl_i32;   // global (device) b32
typedef __attribute__((address_space(3))) int lds_i32;   // LDS b32
#else
#define HAVE_ASYNC_LDS 0
#endif

// ---------------------------------------------------------------------------
// Kernel 1: per-batch top-1500 by radix select + LDS bitonic sort, then decode
// only the selected boxes (avoids materializing the 39MB decoded array).
// One 1024-thread block per batch; conf is L2-resident so 4 histogram passes
// are cheap.
// ---------------------------------------------------------------------------
__global__ __launch_bounds__(1024) void topk_decode_kernel(
    const float* __restrict__ loc, const float* __restrict__ conf,
    const float* __restrict__ prior, float* __restrict__ topk)
{
  const int b   = blockIdx.x;
  const int tid = threadIdx.x;
  const int nt  = blockDim.x;

  __shared__ u32 hist[256];
  __shared__ u32 sh_pref, sh_ca, sh_cnt;
  __shared__ u64 buf[2048];   // ~inverted (key<<32 | (0xFFFFFFFF - idx)); ascending sort == desc key, asc idx

  const float* score_base = conf + (size_t)b * NP * 2 + 1;

  // 4-pass byte radix select: find 32-bit pivot key with count(key > pivot) < TK <= count(key >= pivot).
  u32 prefix = 0, countAbove = 0;
  for (int shift = 24; shift >= 0; shift -= 8) {
    for (int t = tid; t < 256; t += nt) hist[t] = 0;
    __syncthreads();
    const u32 pmask = (shift == 24) ? 0u : (0xFFFFFFFFu << (shift + 8));
    for (int p = tid; p < NP; p += nt) {
      float s = score_base[(size_t)p * 2];
      u32 key = (s > CONF_T) ? __float_as_uint(s) : 0u;  // positive floats: bit order == value order
      if ((key & pmask) == prefix) atomicAdd(&hist[(key >> shift) & 255u], 1u);
    }
    __syncthreads();
    if (tid == 0) {
      u32 remaining = (u32)TK - countAbove;
      u32 cum = 0; int bin = 0;
      for (int bi = 255; bi >= 0; --bi) {
        u32 h = hist[bi];
        if (cum + h >= remaining) { bin = bi; break; }
        cum += h;
      }
      sh_ca   = countAbove + cum;
      sh_pref = prefix | ((u32)bin << shift);
    }
    __syncthreads();
    prefix = sh_pref; countAbove = sh_ca;
    __syncthreads();
  }
  const u32 pivot = prefix;   // pivot==0 -> fewer than TK valid scores (pad with -inf rows)

  // Compact all candidates (key > pivot, plus key == pivot when pivot != 0) into LDS.
  if (tid == 0) sh_cnt = 0;
  __syncthreads();
  for (int p = tid; p < NP; p += nt) {
    float s = score_base[(size_t)p * 2];
    u32 key = (s > CONF_T) ? __float_as_uint(s) : 0u;
    if (key > pivot || (pivot != 0u && key == pivot)) {
      u32 slot = atomicAdd(&sh_cnt, 1u);
      if (slot < 2048u)
        buf[slot] = ~(((u64)key << 32) | (u64)(0xFFFFFFFFu - (u32)p));
    }
  }
  __syncthreads();
  const u32 n = sh_cnt < 2048u ? sh_cnt : 2048u;
  for (int i = tid; i < 2048; i += nt)
    if ((u32)i >= n) buf[i] = ~0ull;       // inv of key64==0 -> sorts to the end
  __syncthreads();

  // Bitonic sort (ascending on inverted keys == descending scores, ties by lower index).
  for (u32 k = 2; k <= 2048u; k <<= 1) {
    for (u32 j = k >> 1; j > 0; j >>= 1) {
      __syncthreads();
      for (u32 i = tid; i < 2048u; i += (u32)nt) {
        u32 ixj = i ^ j;
        if (ixj > i) {
          u64 a = buf[i], c = buf[ixj];
          bool up = ((i & k) == 0);
          if ((a > c) == up) { buf[i] = c; buf[ixj] = a; }
        }
      }
    }
  }
  __syncthreads();

  // Decode only the 1500 winners and write [score,x1,y1,x2,y2] rows.
  float* trow = topk + (size_t)b * TK * 5;
  for (int e = tid; e < TK; e += nt) {
    u64 kv = ~buf[e];
    u32 key = (u32)(kv >> 32);
    float sc, x1, y1, x2, y2;
    if (key == 0u) {
      sc = -INFINITY; x1 = y1 = x2 = y2 = 0.f;
    } else {
      u32 p = 0xFFFFFFFFu - (u32)kv;
      sc = __uint_as_float(key);
      const float4 l  = *(const float4*)(loc + ((size_t)b * NP + p) * 4);
      const float4 pr = *(const float4*)(prior + (size_t)p * 4);
      float cx = pr.x + l.x * 0.1f * pr.z;
      float cy = pr.y + l.y * 0.1f * pr.w;
      float w  = pr.z * expf(l.z * 0.2f);
      float h  = pr.w * expf(l.w * 0.2f);
      x1 = cx - 0.5f * w; y1 = cy - 0.5f * h;
      x2 = x1 + w;        y2 = y1 + h;
    }
    trow[(size_t)e * 5 + 0] = sc;
    trow[(size_t)e * 5 + 1] = x1;
    trow[(size_t)e * 5 + 2] = y1;
    trow[(size_t)e * 5 + 3] = x2;
    trow[(size_t)e * 5 + 4] = y2;
  }
}

// ---------------------------------------------------------------------------
// Kernel 2: parallel suppression matrix. One wave per row i; each 32-bit mask
// word is produced directly by a wave32 ballot. Boxes staged via gfx1250
// async global->LDS loads when available.
// ---------------------------------------------------------------------------
__global__ __launch_bounds__(256) void nms_mask_kernel(
    const float* __restrict__ topk, u32* __restrict__ mask)
{
  const int b   = blockIdx.x;
  const int tid = threadIdx.x;
  __shared__ float sx1[TK], sy1[TK], sx2[TK], sy2[TK], sar[TK];

  const float* trow = topk + (size_t)b * TK * 5;

#if HAVE_ASYNC_LDS
  for (int t = tid; t < TK; t += 256) {
    const float* g = trow + (size_t)t * 5;
    __builtin_amdgcn_global_load_async_to_lds_b32(
        (gbl_i32*)(g + 1), (lds_i32*)&sx1[t], 0, 0);
    __builtin_amdgcn_global_load_async_to_lds_b32(
        (gbl_i32*)(g + 2), (lds_i32*)&sy1[t], 0, 0);
    __builtin_amdgcn_global_load_async_to_lds_b32(
        (gbl_i32*)(g + 3), (lds_i32*)&sx2[t], 0, 0);
    __builtin_amdgcn_global_load_async_to_lds_b32(
        (gbl_i32*)(g + 4), (lds_i32*)&sy2[t], 0, 0);
  }
  __builtin_amdgcn_s_wait_asynccnt(0);
  __syncthreads();
  for (int t = tid; t < TK; t += 256)
    sar[t] = (sx2[t] - sx1[t]) * (sy2[t] - sy1[t]);
  __syncthreads();
#else
  for (int t = tid; t < TK; t += 256) {
    float x1 = trow[(size_t)t * 5 + 1], y1 = trow[(size_t)t * 5 + 2];
    float x2 = trow[(size_t)t * 5 + 3], y2 = trow[(size_t)t * 5 + 4];
    sx1[t] = x1; sy1[t] = y1; sx2[t] = x2; sy2[t] = y2;
    sar[t] = (x2 - x1) * (y2 - y1);
  }
  __syncthreads();
#endif

  const int wid = tid >> 5, lane = tid & 31;
  const int i = blockIdx.y * 8 + wid;       // uniform per wave
  if (i >= TK) return;
  const float bx1 = sx1[i], by1 = sy1[i], bx2 = sx2[i], by2 = sy2[i], bar = sar[i];
  u32* row = mask + ((size_t)b * TK + i) * MASKW;

  for (int w = 0; w < MASKW - 1; ++w) {
    int j = w * 32 + lane;
    bool sup = false;
    if (j < TK && j != i) {
      float xx1 = fmaxf(sx1[j], bx1);
      float yy1 = fmaxf(sy1[j], by1);
      float xx2 = fminf(sx2[j], bx2);
      float yy2 = fminf(sy2[j], by2);
      float inter = fmaxf(xx2 - xx1, 0.f) * fmaxf(yy2 - yy1, 0.f);
      float iou = inter / (sar[j] + bar - inter);   // 0/0 -> NaN -> compare false (matches padded boxes)
      sup = iou > NMS_T;
    }
    u32 m = ballot32(sup);
    if (lane == 0) row[w] = m;
  }
  if (lane == 0) row[MASKW - 1] = 0u;
}

// ---------------------------------------------------------------------------
// Kernel 3: serial greedy scan, one wave per batch. 48 active words in LDS;
// per kept box: 48-word AND of its L2-resident mask row. Single-wave
// workgroup -> barriers are S_NOPs per the ISA.
// ---------------------------------------------------------------------------
__global__ __launch_bounds__(32) void nms_scan_kernel(
    const float* __restrict__ topk, const u32* __restrict__ mask,
    u32* __restrict__ actout)
{
  const int b = blockIdx.x;
  const int lane = threadIdx.x;
  __shared__ u32 act[MASKW];

  const float* trow = topk + (size_t)b * TK * 5;
  for (int w = 0; w < MASKW; ++w) {
    int j = w * 32 + lane;
    bool v = (j < TK) && (trow[(size_t)j * 5] > -1e30f);   // valid == score above -inf padding
    u32 m = ballot32(v);
    if (lane == 0) act[w] = m;
  }
  __syncthreads();

  const u32* mbase = mask + (size_t)b * TK * MASKW;
  for (int i = 0; i < TK; ++i) {
    u32 word = act[i >> 5];                 // same-address LDS read: broadcast
    if ((word >> (i & 31)) & 1u) {
      const u32* row = mbase + (size_t)i * MASKW;
      u32 m0 = row[lane];
      act[lane] &= ~m0;                     // bit i survives: row excludes j==i
      if (lane < 16) {
        u32 m1 = row[lane + 32];
        act[lane + 32] &= ~m1;
      }
    }
    __syncthreads();
  }
  actout[b * MASKW + lane] = act[lane];
  if (lane < 16) actout[b * MASKW + lane + 32] = act[lane + 32];
}

// ---------------------------------------------------------------------------
// Kernel 4: compaction ranks + output. Exclusive prefix scan of the 48 word
// popcounts is done with the matrix engine: group the 48 words into 16 sums,
// multiply by a strict-lower-triangular ones matrix via 4 chained
// v_wmma_f32_16x16x4_f32 (K split 16 = 4x4), per the §7.12.2 f32 layouts.
// ---------------------------------------------------------------------------
__global__ __launch_bounds__(256) void output_kernel(
    const float* __restrict__ topk, const u32* __restrict__ actin,
    float* __restrict__ out)
{
  const int b = blockIdx.x;
  const int tid = threadIdx.x;
  __shared__ u32  aw[MASKW];
  __shared__ u32  pc[MASKW];
  __shared__ float gsum[16];
  __shared__ float gexc[16];

  float* ob = out + (size_t)b * 2 * TK * 5;
  for (int t = tid; t < 2 * TK * 5; t += 256) ob[t] = 0.f;  // background class + unfilled rows

  if (tid < MASKW) {
    u32 a = actin[b * MASKW + tid];
    aw[tid] = a;
    pc[tid] = (u32)__popc(a);
  }
  __syncthreads();
  if (tid < 16) gsum[tid] = (float)(pc[3 * tid] + pc[3 * tid + 1] + pc[3 * tid + 2]);
  __syncthreads();

#if __has_builtin(__builtin_amdgcn_wmma_f32_16x16x4_f32)
  if (tid < 32) {                 // exactly wave 0, EXEC all-ones (WMMA requirement)
    const int lane = tid;
    const int M    = lane & 15;   // A row / D column
    const int koff = (lane >> 4) * 2;
    v8f acc = {0.f, 0.f, 0.f, 0.f, 0.f, 0.f, 0.f, 0.f};
#pragma unroll
    for (int k0 = 0; k0 < 16; k0 += 4) {
      const int ka = k0 + koff;   // A 16x4 f32 layout: v0 holds K=ka, v1 holds K=ka+1
      v2f a, bb;
      a.x = (ka     < M) ? 1.f : 0.f;   // strict lower-triangular ones -> exclusive scan
      a.y = (ka + 1 < M) ? 1.f : 0.f;
      bb.x = gsum[ka];                  // B 4x16: row K broadcast across all 16 columns
      bb.y = gsum[ka + 1];
      acc = __builtin_amdgcn_wmma_f32_16x16x4_f32(false, a, false, bb,
                                                  (short)0, acc, false, false);
    }
    // D layout: lane 0 = col 0 rows 0..7, lane 16 = col 0 rows 8..15.
    if (lane == 0) {
#pragma unroll
      for (int v = 0; v < 8; ++v) gexc[v] = acc[v];
    }
    if (lane == 16) {
#pragma unroll
      for (int v = 0; v < 8; ++v) gexc[8 + v] = acc[v];
    }
  }
#else
  if (tid == 0) { float s = 0.f; for (int t = 0; t < 16; ++t) { gexc[t] = s; s += gsum[t]; } }
#endif
  __syncthreads();

  const float* trow = topk + (size_t)b * TK * 5;
  for (int j = tid; j < TK; j += 256) {
    const int wi = j >> 5, bit = j & 31;
    if ((aw[wi] >> bit) & 1u) {
      const int gq = wi / 3, r = wi - gq * 3;
      u32 rank = (u32)(gexc[gq] + 0.5f);
      if (r > 0) rank += pc[3 * gq];
      if (r > 1) rank += pc[3 * gq + 1];
      rank += (u32)__popc(aw[wi] & ((1u << bit) - 1u));
      float* dst = ob + ((size_t)TK + rank) * 5;   // class 1 slab
      dst[0] = trow[(size_t)j * 5 + 0];
      dst[1] = trow[(size_t)j * 5 + 1];
      dst[2] = trow[(size_t)j * 5 + 2];
      dst[3] = trow[(size_t)j * 5 + 3];
      dst[4] = trow[(size_t)j * 5 + 4];
    }
  }
}

// ---------------------------------------------------------------------------
extern "C" void kernel_launch(void* const* d_in, const int* in_sizes, int n_in,
                              void* d_out, int out_size, void* d_ws, size_t ws_size,
                              hipStream_t stream)
{
  (void)in_sizes; (void)n_in; (void)out_size; (void)ws_size;
  const float* loc   = (const float*)d_in[0];
  const float* conf  = (const float*)d_in[1];
  const float* prior = (const float*)d_in[2];
  float* out = (float*)d_out;

  char* ws = (char*)d_ws;
  float* ws_topk = (float*)ws;                                            // 32*1500*5 f32
  u32*   ws_mask = (u32*)(ws + (size_t)NB * TK * 5 * sizeof(float));      // 32*1500*48 u32
  u32*   ws_act  = (u32*)(ws + (size_t)NB * TK * 5 * sizeof(float)
                             + (size_t)NB * TK * MASKW * sizeof(u32));    // 32*48 u32

  topk_decode_kernel<<<NB, 1024, 0, stream>>>(loc, conf, prior, ws_topk);
  nms_mask_kernel<<<dim3(NB, (TK + 7) / 8), 256, 0, stream>>>(ws_topk, ws_mask);
  nms_scan_kernel<<<NB, 32, 0, stream>>>(ws_topk, ws_mask, ws_act);
  output_kernel<<<NB, 256, 0, stream>>>(ws_topk, ws_act, out);
}